// NGMDA_70592082477288
// MI455X (gfx1250) — compile-verified
//
#include <hip/hip_runtime.h>

typedef float v2f __attribute__((ext_vector_type(2)));
typedef float v8f __attribute__((ext_vector_type(8)));

#define WMMA_F32(a, b, c) \
  __builtin_amdgcn_wmma_f32_16x16x4_f32(false, (a), false, (b), (short)0, (c), false, false)

constexpr int   kNDrug = 1373;
constexpr int   kNodes = 1546;
constexpr int   kInF   = 1546;
constexpr int   kOut   = 64;
constexpr int   kHeads = 8;
constexpr int   kHO    = 512;     // HEADS*OUT
constexpr float kTao   = 0.4f;
constexpr float kEps   = 1e-12f;

__device__ __forceinline__ float waveRedSum(float v) {
#pragma unroll
  for (int m = 16; m; m >>= 1) v += __shfl_xor(v, m, 32);
  return v;
}
__device__ __forceinline__ float waveRedMax(float v) {
#pragma unroll
  for (int m = 16; m; m >>= 1) v = fmaxf(v, __shfl_xor(v, m, 32));
  return v;
}
// monotonic float <-> uint encoding so atomicMax(uint) == float max
__device__ __forceinline__ unsigned encF(float f) {
  unsigned u = __float_as_uint(f);
  return (u & 0x80000000u) ? ~u : (u | 0x80000000u);
}
__device__ __forceinline__ float decF(unsigned u) {
  return (u & 0x80000000u) ? __uint_as_float(u & 0x7FFFFFFFu) : __uint_as_float(~u);
}

// ---------------------------------------------------------------------------
// Kernel 1: feat @ W_proj[sel] and feat @ W_res[sel] (sel = drug/non-drug rows)
// grid.x: 32-row blocks (49), grid.z: 0 -> W_proj/feat_src, 1 -> W_res/resval
// block 256 = 8 waves; wave w covers columns [w*64, w*64+64)
// ---------------------------------------------------------------------------
__global__ void __launch_bounds__(256)
k_proj_gemm(const float* __restrict__ feat, const float* __restrict__ Wp,
            const float* __restrict__ Wr, float* __restrict__ feat_src,
            float* __restrict__ resval) {
  const int lane = threadIdx.x & 31;
  const int wv = threadIdx.x >> 5;
  const int rowBase = blockIdx.x * 32;
  const int colBase = wv * 64;
  const float* Wsel = blockIdx.z ? Wr : Wp;
  float* outp = blockIdx.z ? resval : feat_src;
  const int nl = lane & 15;
  const int kk = (lane >> 4) * 2;
  const int mHi = (lane >> 4) * 8;

  int r0 = rowBase + nl;       if (r0 > kNodes - 1) r0 = kNodes - 1;
  int r1 = rowBase + 16 + nl;  if (r1 > kNodes - 1) r1 = kNodes - 1;
  const float* a0p = feat + (size_t)r0 * kInF;
  const float* a1p = feat + (size_t)r1 * kInF;

  for (int pass = 0; pass < 2; ++pass) {
    const int segLo = pass ? kNDrug : 0;
    const int segHi = pass ? kNodes : kNDrug;
    if (rowBase >= segHi || rowBase + 32 <= segLo) continue;
    const float* W = Wsel + (size_t)pass * kInF * kHO;

    v8f acc[2][4];
#pragma unroll
    for (int r = 0; r < 2; ++r)
#pragma unroll
      for (int j = 0; j < 4; ++j) acc[r][j] = {};

    int k0 = 0;
    for (; k0 + 4 <= kInF; k0 += 4) {
      v2f a0 = *(const v2f*)(a0p + k0 + kk);
      v2f a1 = *(const v2f*)(a1p + k0 + kk);
      const float* wr0 = W + (size_t)(k0 + kk) * kHO + colBase + nl;
      const float* wr1 = wr0 + kHO;
      v2f b[4];
#pragma unroll
      for (int j = 0; j < 4; ++j) { b[j].x = wr0[j * 16]; b[j].y = wr1[j * 16]; }
#pragma unroll
      for (int j = 0; j < 4; ++j) {
        acc[0][j] = WMMA_F32(a0, b[j], acc[0][j]);
        acc[1][j] = WMMA_F32(a1, b[j], acc[1][j]);
      }
    }
    if (k0 < kInF) {  // tail: K%4 == 2, only kk==0 half-wave has valid data
      const bool lo = (kk == 0);
      v2f a0 = {}, a1 = {};
      v2f b[4]; 
#pragma unroll
      for (int j = 0; j < 4; ++j) b[j] = {};
      if (lo) {
        a0 = *(const v2f*)(a0p + k0);
        a1 = *(const v2f*)(a1p + k0);
        const float* wr0 = W + (size_t)k0 * kHO + colBase + nl;
        const float* wr1 = wr0 + kHO;
#pragma unroll
        for (int j = 0; j < 4; ++j) { b[j].x = wr0[j * 16]; b[j].y = wr1[j * 16]; }
      }
#pragma unroll
      for (int j = 0; j < 4; ++j) {
        acc[0][j] = WMMA_F32(a0, b[j], acc[0][j]);
        acc[1][j] = WMMA_F32(a1, b[j], acc[1][j]);
      }
    }
#pragma unroll
    for (int r = 0; r < 2; ++r)
#pragma unroll
      for (int j = 0; j < 4; ++j)
#pragma unroll
        for (int i = 0; i < 8; ++i) {
          const int row = rowBase + r * 16 + mHi + i;
          if (row >= segLo && row < segHi)
            outp[(size_t)row * kHO + colBase + j * 16 + nl] = acc[r][j][i];
        }
  }
}

// ---------------------------------------------------------------------------
// Kernel 2: logits[n,h,:] = tanh(feat_src[n,h,:] @ W_prob[h])   (K = N = 64)
// grid.x: 16-row tiles (97); block 256, wave w -> head w
// ---------------------------------------------------------------------------
__global__ void __launch_bounds__(256)
k_prob_gemm_tanh(const float* __restrict__ feat_src, const float* __restrict__ Wprob,
                 float* __restrict__ logits) {
  const int lane = threadIdx.x & 31;
  const int h = threadIdx.x >> 5;
  const int rowBase = blockIdx.x * 16;
  const int nl = lane & 15;
  const int kk = (lane >> 4) * 2;
  const int mHi = (lane >> 4) * 8;

  int rr = rowBase + nl; if (rr > kNodes - 1) rr = kNodes - 1;
  const float* ap = feat_src + (size_t)rr * kHO + h * kOut;
  const float* W = Wprob + (size_t)h * kOut * kOut;

  v8f acc[4];
#pragma unroll
  for (int j = 0; j < 4; ++j) acc[j] = {};

  for (int k0 = 0; k0 < kOut; k0 += 4) {
    v2f a = *(const v2f*)(ap + k0 + kk);
    const float* w0 = W + (size_t)(k0 + kk) * kOut + nl;
    const float* w1 = w0 + kOut;
    v2f b[4];
#pragma unroll
    for (int j = 0; j < 4; ++j) { b[j].x = w0[j * 16]; b[j].y = w1[j * 16]; }
#pragma unroll
    for (int j = 0; j < 4; ++j) acc[j] = WMMA_F32(a, b[j], acc[j]);
  }
#pragma unroll
  for (int j = 0; j < 4; ++j)
#pragma unroll
    for (int i = 0; i < 8; ++i) {
      const int row = rowBase + mHi + i;
      if (row < kNodes)
        logits[(size_t)row * kHO + h * kOut + j * 16 + nl] = tanhf(acc[j][i]);
    }
}

// Kernel 3: in-place softmax over rows of 64 (N*H rows); one wave per row
__global__ void __launch_bounds__(256)
k_softmax64(float* __restrict__ logits, int rows) {
  const int lane = threadIdx.x & 31;
  const int r = blockIdx.x * 8 + (threadIdx.x >> 5);
  if (r >= rows) return;
  float* p = logits + (size_t)r * 64;
  v2f v = *(const v2f*)(p + 2 * lane);
  const float mx = waveRedMax(fmaxf(v.x, v.y));
  const float e0 = expf(v.x - mx), e1 = expf(v.y - mx);
  const float inv = 1.0f / waveRedSum(e0 + e1);
  v2f o; o.x = e0 * inv; o.y = e1 * inv;
  *(v2f*)(p + 2 * lane) = o;
}

// Kernel 4: column (axis-0) inverse norms of feat_src (512 columns)
__global__ void k_colnorm(const float* __restrict__ fs, float* __restrict__ inv_norm) {
  const int c = blockIdx.x * 256 + threadIdx.x;
  if (c >= kHO) return;
  float s = 0.0f;
  for (int n = 0; n < kNodes; ++n) { float x = fs[(size_t)n * kHO + c]; s += x * x; }
  inv_norm[c] = 1.0f / fmaxf(sqrtf(s), kEps);
}

// Kernel 5: ft_norm = feat_src * inv_norm[col]
__global__ void k_scale_cols(const float* __restrict__ fs, const float* __restrict__ inv_norm,
                             float* __restrict__ ftn, int total) {
  const int i = blockIdx.x * 256 + threadIdx.x;
  if (i < total) ftn[i] = fs[i] * inv_norm[i & (kHO - 1)];
}

// Kernel 6: per-node L2-normalized sloc/topo rows; grid.y selects array
__global__ void __launch_bounds__(256)
k_rownorm64(const float* __restrict__ sloc, const float* __restrict__ topo,
            float* __restrict__ sloc_n, float* __restrict__ topo_n) {
  const int lane = threadIdx.x & 31;
  const int n = blockIdx.x * 8 + (threadIdx.x >> 5);
  if (n >= kNodes) return;
  const float* in = blockIdx.y ? topo : sloc;
  float* out = blockIdx.y ? topo_n : sloc_n;
  v2f v = *(const v2f*)(in + (size_t)n * 64 + 2 * lane);
  const float s = waveRedSum(v.x * v.x + v.y * v.y);
  const float inv = 1.0f / fmaxf(sqrtf(s), kEps);
  v2f o; o.x = v.x * inv; o.y = v.y * inv;
  *(v2f*)(out + (size_t)n * 64 + 2 * lane) = o;
}

__global__ void k_zero(unsigned* __restrict__ p, int n) {
  const int i = blockIdx.x * 256 + threadIdx.x;
  if (i < n) p[i] = 0u;
}

// Kernel 7: per edge -> unnormalized logits + segment-max; one wave per edge
__global__ void __launch_bounds__(256)
k_edge_logits(const int* __restrict__ src, const int* __restrict__ dst,
              const int* __restrict__ efeat, const float* __restrict__ etattn,
              const float* __restrict__ sloc_n, const float* __restrict__ topo_n,
              const float* __restrict__ ft_attn, const float* __restrict__ ft_norm,
              float* __restrict__ exbuf, unsigned* __restrict__ emax, int E) {
  const int lane = threadIdx.x & 31;
  const int e = blockIdx.x * 8 + (threadIdx.x >> 5);
  if (e >= E) return;
  const int s = src[e], t = dst[e];

  v2f a = *(const v2f*)(sloc_n + (size_t)s * 64 + 2 * lane);
  v2f b = *(const v2f*)(sloc_n + (size_t)t * 64 + 2 * lane);
  const float cs = waveRedSum(a.x * b.x + a.y * b.y);
  a = *(const v2f*)(topo_n + (size_t)s * 64 + 2 * lane);
  b = *(const v2f*)(topo_n + (size_t)t * 64 + 2 * lane);
  const float ct = waveRedSum(a.x * b.x + a.y * b.y);
  const float sim = 0.5f * (kTao * cs + (1.0f - kTao) * ct + 1.0f);
  const float scale = sim * etattn[efeat[e]];

  const float* pa = ft_attn + (size_t)t * kHO;
  const float* pn = ft_norm + (size_t)s * kHO;
#pragma unroll
  for (int h = 0; h < kHeads; ++h) {
    v2f fa = *(const v2f*)(pa + h * 64 + 2 * lane);
    v2f fn = *(const v2f*)(pn + h * 64 + 2 * lane);
    const float lg = scale * waveRedSum(fa.x * fn.x + fa.y * fn.y);
    if (lane == h) exbuf[(size_t)e * 8 + h] = lg;
    if (lane == 16 + h) atomicMax(emax + t * 8 + h, encF(lg));
  }
}

// Kernel 8: ex = exp(logit - segmax); segment-sum denom
__global__ void k_edge_exp(const int* __restrict__ dst, float* __restrict__ exbuf,
                           const unsigned* __restrict__ emax, float* __restrict__ denom,
                           int E8) {
  const int i = blockIdx.x * 256 + threadIdx.x;
  if (i >= E8) return;
  const int e = i >> 3, h = i & 7;
  const int t = dst[e];
  const float ex = expf(exbuf[i] - decF(emax[t * 8 + h]));
  exbuf[i] = ex;
  atomicAdd(denom + t * 8 + h, ex);
}

// Kernel 9: att = ex/denom; scatter-add rst / sagg / tagg; one wave per edge
__global__ void __launch_bounds__(256)
k_edge_scatter(const int* __restrict__ src, const int* __restrict__ dst,
               const float* __restrict__ exbuf, const float* __restrict__ denom,
               const float* __restrict__ feat_src, const float* __restrict__ sloc,
               const float* __restrict__ topo, float* __restrict__ rst,
               float* __restrict__ sagg, float* __restrict__ tagg, int E) {
  const int lane = threadIdx.x & 31;
  const int e = blockIdx.x * 8 + (threadIdx.x >> 5);
  if (e >= E) return;
  const int s = src[e], t = dst[e];
  const float* fsrc = feat_src + (size_t)s * kHO;
  float* rdst = rst + (size_t)t * kHO;
  float asum = 0.0f;
#pragma unroll
  for (int h = 0; h < kHeads; ++h) {
    const float a_h = exbuf[(size_t)e * 8 + h] / denom[t * 8 + h];
    asum += a_h;
    const int o = h * 64 + lane;
    atomicAdd(rdst + o, a_h * fsrc[o]);
    atomicAdd(rdst + o + 32, a_h * fsrc[o + 32]);
  }
  atomicAdd(sagg + (size_t)t * 64 + lane, asum * sloc[(size_t)s * 64 + lane]);
  atomicAdd(sagg + (size_t)t * 64 + lane + 32, asum * sloc[(size_t)s * 64 + lane + 32]);
  atomicAdd(tagg + (size_t)t * 64 + lane, asum * topo[(size_t)s * 64 + lane]);
  atomicAdd(tagg + (size_t)t * 64 + lane + 32, asum * topo[(size_t)s * 64 + lane + 32]);
}

// Kernel 10: finalize all three outputs
__global__ void k_finalize(const float* __restrict__ rst, const float* __restrict__ resval,
                           const float* __restrict__ sagg, const float* __restrict__ tagg,
                           const float* __restrict__ sloc, const float* __restrict__ topo,
                           float* __restrict__ out) {
  const int i = blockIdx.x * 256 + threadIdx.x;
  if (i >= kNodes * kOut) return;
  const int n = i >> 6, d = i & 63;
  float sacc = 0.0f;
#pragma unroll
  for (int h = 0; h < kHeads; ++h)
    sacc += fmaxf(rst[(size_t)n * kHO + h * 64 + d] + resval[(size_t)n * kHO + h * 64 + d], 0.0f);
  out[i] = sacc * 0.125f;
  out[kNodes * kOut + i] = sloc[i] + sagg[i] * 0.125f;
  out[2 * kNodes * kOut + i] = topo[i] + tagg[i] * 0.125f;
}

extern "C" void kernel_launch(void* const* d_in, const int* in_sizes, int n_in,
                              void* d_out, int out_size, void* d_ws, size_t ws_size,
                              hipStream_t stream) {
  (void)n_in; (void)out_size; (void)ws_size;
  const float* feat  = (const float*)d_in[0];
  const float* sloc  = (const float*)d_in[1];
  const float* topo  = (const float*)d_in[2];
  const float* Wproj = (const float*)d_in[3];
  const float* Wres  = (const float*)d_in[4];
  const float* Wprob = (const float*)d_in[5];
  const float* etat  = (const float*)d_in[6];
  const int*   src   = (const int*)d_in[7];
  const int*   dst   = (const int*)d_in[8];
  const int*   efeat = (const int*)d_in[9];
  const int E = in_sizes[7];

  const size_t NH  = (size_t)kNodes * kHO;   // 791552
  const size_t N64 = (size_t)kNodes * kOut;  // 98944
  float* ws = (float*)d_ws;
  float* feat_src = ws; ws += NH;
  float* resval   = ws; ws += NH;
  float* logits   = ws; ws += NH;            // becomes ft_attn after softmax
  float* ft_norm  = ws; ws += NH;
  float* inv_norm = ws; ws += kHO;
  float* sloc_n   = ws; ws += N64;
  float* topo_n   = ws; ws += N64;
  float* exbuf    = ws; ws += (size_t)E * 8;
  float* rst      = ws; ws += NH;            // --- zeroed region start
  float* sagg     = ws; ws += N64;
  float* tagg     = ws; ws += N64;
  float* denom    = ws; ws += (size_t)kNodes * kHeads;
  unsigned* emax  = (unsigned*)ws;           // encF init: 0 == lowest
  const int zeroWords = (int)(NH + 2 * N64 + 2 * (size_t)kNodes * kHeads);

  const dim3 blk(256);
  k_zero<<<dim3((zeroWords + 255) / 256), blk, 0, stream>>>((unsigned*)rst, zeroWords);
  k_proj_gemm<<<dim3(49, 1, 2), blk, 0, stream>>>(feat, Wproj, Wres, feat_src, resval);
  k_prob_gemm_tanh<<<dim3(97), blk, 0, stream>>>(feat_src, Wprob, logits);
  k_softmax64<<<dim3((kNodes * kHeads + 7) / 8), blk, 0, stream>>>(logits, kNodes * kHeads);
  k_colnorm<<<dim3(2), blk, 0, stream>>>(feat_src, inv_norm);
  k_scale_cols<<<dim3((int)((NH + 255) / 256)), blk, 0, stream>>>(feat_src, inv_norm, ft_norm, (int)NH);
  k_rownorm64<<<dim3((kNodes + 7) / 8, 2), blk, 0, stream>>>(sloc, topo, sloc_n, topo_n);
  k_edge_logits<<<dim3((E + 7) / 8), blk, 0, stream>>>(src, dst, efeat, etat, sloc_n, topo_n,
                                                       logits, ft_norm, exbuf, emax, E);
  k_edge_exp<<<dim3((E * 8 + 255) / 256), blk, 0, stream>>>(dst, exbuf, emax, denom, E * 8);
  k_edge_scatter<<<dim3((E + 7) / 8), blk, 0, stream>>>(src, dst, exbuf, denom, feat_src,
                                                        sloc, topo, rst, sagg, tagg, E);
  k_finalize<<<dim3((int)((N64 + 255) / 256)), blk, 0, stream>>>(rst, resval, sagg, tagg,
                                                                 sloc, topo, (float*)d_out);
}